// LSTM_28363964023398
// MI455X (gfx1250) — compile-verified
//
#include <hip/hip_runtime.h>

typedef __attribute__((ext_vector_type(2))) float v2f;
typedef __attribute__((ext_vector_type(8))) float v8f;

#define Bsz  32
#define Tlen 50
#define Idim 256
#define Hdim 512
#define G4H  2048   // 4*H

__device__ __forceinline__ float sigf(float x) {
    return 1.0f / (1.0f + __expf(-x));
}

// ---------------------------------------------------------------------------
// Kernel 1: wx[b][t][k] = sum_i Wih[b][k][i] * emb[ids[b][t]][i]
// 32 GEMMs of [2048 x 256] x [256 x 50] using V_WMMA_F32_16X16X4_F32.
// Tile grid: b(32) x mtile(128) x ntile(4, N padded 50->64), one wave per tile.
// ---------------------------------------------------------------------------
__global__ __launch_bounds__(256)
void wx_gemm(const int* __restrict__ ids, const float* __restrict__ emb,
             const float* __restrict__ Wih, float* __restrict__ wx) {
    const int lane = threadIdx.x & 31;
    const int wave = threadIdx.x >> 5;
    const int tile = blockIdx.x * 8 + wave;      // 16384 tiles total
    const int b    = tile >> 9;                  // / (128*4)
    const int rem  = tile & 511;
    const int M0   = (rem >> 2) << 4;            // mtile*16
    const int N0   = (rem & 3) << 4;             // ntile*16
    const int half = lane >> 4;
    const int l15  = lane & 15;

    const int mrow = M0 + l15;                   // A row this lane supplies
    const int tcol = N0 + l15;                   // C column this lane holds
    const int tc   = tcol < Tlen ? tcol : (Tlen - 1);
    const int id   = ids[b * Tlen + tc];

    // A 16x4 f32 layout: lanes<16 hold K={0,1}, lanes>=16 hold K={2,3}
    const float* arow = Wih + ((size_t)b * G4H + mrow) * Idim + 2 * half;
    // B 4x16 f32 layout mirrored: lanes<16 hold K={0,1} of column tcol
    const float* brow = emb + (size_t)id * Idim + 2 * half;

    v8f acc = {};
    #pragma unroll 4
    for (int kk = 0; kk < Idim; kk += 4) {
        v2f a, bv;
        a.x  = arow[kk + 0];
        a.y  = arow[kk + 1];
        bv.x = brow[kk + 0];
        bv.y = brow[kk + 1];
        acc = __builtin_amdgcn_wmma_f32_16x16x4_f32(
            /*neg_a=*/false, a, /*neg_b=*/false, bv,
            /*c_mod=*/(short)0, acc, /*reuse_a=*/false, /*reuse_b=*/false);
    }

    if (tcol < Tlen) {
        // C layout: VGPR r holds (M = M0 + r + 8*half, N = tcol)
        float* orow = wx + ((size_t)(b * Tlen + tcol)) * G4H;
        #pragma unroll
        for (int r = 0; r < 8; ++r)
            orow[M0 + r + 8 * half] = acc[r];
    }
}

// ---------------------------------------------------------------------------
// Kernel 2: persistent LSTM recurrence. One workgroup per sample (independent
// recurrences -> no cross-WG sync). h lives in LDS, c in registers. Each
// thread owns h-rows {tid, tid+256} across the 4 gates (8 dot products/step).
// weight_hh[b] (4MB) streams from L2 each of the 50 steps.
// ---------------------------------------------------------------------------
__global__ __launch_bounds__(256)
void lstm_recur(const float* __restrict__ Whh, const float* __restrict__ bih,
                const float* __restrict__ bhh, const float* __restrict__ wx,
                float* __restrict__ all_hidden) {
    __shared__ __align__(16) float hs[Hdim];
    const int b   = blockIdx.x;
    const int tid = threadIdx.x;
    const int r0  = tid;
    const int r1  = tid + 256;

    const float* W   = Whh + (size_t)b * G4H * Hdim;
    const float* wxb = wx + (size_t)b * Tlen * G4H;

    float bias[8];
    const float* Wrow[8];
    #pragma unroll
    for (int q = 0; q < 4; ++q) {
        const int ka = q * Hdim + r0;
        const int kb = q * Hdim + r1;
        bias[q * 2 + 0] = bih[b * G4H + ka] + bhh[b * G4H + ka];
        bias[q * 2 + 1] = bih[b * G4H + kb] + bhh[b * G4H + kb];
        Wrow[q * 2 + 0] = W + (size_t)ka * Hdim;
        Wrow[q * 2 + 1] = W + (size_t)kb * Hdim;
    }

    float c0 = 0.f, c1 = 0.f;
    hs[r0] = 0.f;
    hs[r1] = 0.f;
    __syncthreads();

    for (int t = 0; t < Tlen; ++t) {
        const float* wxt = wxb + (size_t)t * G4H;
        if (t + 1 < Tlen)  // global_prefetch_b8 for next step's wx slice
            __builtin_prefetch(wxb + (size_t)(t + 1) * G4H + r0, 0, 0);

        float acc[8];
        #pragma unroll
        for (int q = 0; q < 4; ++q) {
            acc[q * 2 + 0] = bias[q * 2 + 0] + wxt[q * Hdim + r0];
            acc[q * 2 + 1] = bias[q * 2 + 1] + wxt[q * Hdim + r1];
        }

        for (int kk = 0; kk < Hdim; kk += 4) {
            const float4 hv = *(const float4*)&hs[kk];   // ds_load_b128
            #pragma unroll
            for (int j = 0; j < 8; ++j) {
                const float4 w = *(const float4*)(Wrow[j] + kk);
                acc[j] = fmaf(w.x, hv.x, acc[j]);
                acc[j] = fmaf(w.y, hv.y, acc[j]);
                acc[j] = fmaf(w.z, hv.z, acc[j]);
                acc[j] = fmaf(w.w, hv.w, acc[j]);
            }
        }

        const float i0 = sigf(acc[0]), i1 = sigf(acc[1]);
        const float f0 = sigf(acc[2]), f1 = sigf(acc[3]);
        const float o0 = sigf(acc[4]), o1 = sigf(acc[5]);
        const float g0 = tanhf(acc[6]), g1 = tanhf(acc[7]);

        c0 = f0 * c0 + i0 * g0;
        c1 = f1 * c1 + i1 * g1;
        const float h0 = o0 * fmaxf(c0, 0.f);   // relu(c), per reference
        const float h1 = o1 * fmaxf(c1, 0.f);

        __syncthreads();                         // all dot-reads of hs done
        hs[r0] = h0;
        hs[r1] = h1;
        float* hout = all_hidden + ((size_t)(b * Tlen + t)) * Hdim;
        hout[r0] = h0;
        hout[r1] = h1;
        __syncthreads();                         // hs visible for next step
    }
}

// ---------------------------------------------------------------------------
// Kernel 3: all_outputs[b,t,:] = h(b,t) . w_lin^T + b_lin ; one wave per (b,t).
// softmax_out[b] = last-step row (hT @ w_lin^T + b_lin == all_outputs[b,49,:]).
// ---------------------------------------------------------------------------
__global__ __launch_bounds__(32)
void proj(const float* __restrict__ all_hidden, const float* __restrict__ wlin,
          const float* __restrict__ blin, float* __restrict__ out_sm,
          float* __restrict__ out_all) {
    const int bt   = blockIdx.x;    // b*Tlen + t
    const int lane = threadIdx.x;
    const float* h = all_hidden + (size_t)bt * Hdim;

    float s0 = 0.f, s1 = 0.f;
    for (int k = lane; k < Hdim; k += 32) {
        const float hv = h[k];
        s0 = fmaf(hv, wlin[k], s0);
        s1 = fmaf(hv, wlin[Hdim + k], s1);
    }
    #pragma unroll
    for (int off = 16; off > 0; off >>= 1) {
        s0 += __shfl_down(s0, off, 32);
        s1 += __shfl_down(s1, off, 32);
    }
    if (lane == 0) {
        const int t = bt % Tlen;
        const int b = bt / Tlen;
        const float y0 = s0 + blin[0];
        const float y1 = s1 + blin[1];
        out_all[bt * 2 + 0] = y0;
        out_all[bt * 2 + 1] = y1;
        if (t == Tlen - 1) {
            out_sm[b * 2 + 0] = y0;
            out_sm[b * 2 + 1] = y1;
        }
    }
}

extern "C" void kernel_launch(void* const* d_in, const int* in_sizes, int n_in,
                              void* d_out, int out_size, void* d_ws, size_t ws_size,
                              hipStream_t stream) {
    const int*   ids  = (const int*)d_in[0];     // [32,50]
    const float* emb  = (const float*)d_in[1];   // [32000,256]
    const float* Wih  = (const float*)d_in[2];   // [32,2048,256]
    const float* Whh  = (const float*)d_in[3];   // [32,2048,512]
    const float* bih  = (const float*)d_in[4];   // [32,2048]
    const float* bhh  = (const float*)d_in[5];   // [32,2048]
    const float* wlin = (const float*)d_in[6];   // [2,512]
    const float* blin = (const float*)d_in[7];   // [2]

    float* out        = (float*)d_out;
    float* out_sm     = out;                           // [32,2]
    float* all_hidden = out + Bsz * 2;                 // [32,50,512]
    float* out_all    = out + Bsz * 2 + Bsz * Tlen * Hdim;  // [32,50,2]

    float* wx = (float*)d_ws;                          // [32,50,2048] = 13.1 MB

    wx_gemm<<<2048, 256, 0, stream>>>(ids, emb, Wih, wx);
    lstm_recur<<<Bsz, 256, 0, stream>>>(Whh, bih, bhh, wx, all_hidden);
    proj<<<Bsz * Tlen, 32, 0, stream>>>(all_hidden, wlin, blin, out_sm, out_all);
}